// YOLOv3_60301340836035
// MI455X (gfx1250) — compile-verified
//
#include <hip/hip_runtime.h>

typedef float v2f __attribute__((ext_vector_type(2)));
typedef float v8f __attribute__((ext_vector_type(8)));

__device__ __forceinline__ float sigm(float x) {
  return 1.0f / (1.0f + __expf(-x));
}

// Sum a per-lane float across the wave32 using V_WMMA_F32_16X16X4_F32 with
// B = all-ones (layout-independent). D = A x ones: every column of D holds the
// 16 row-sums of A. Per-lane sum of the 8 D VGPRs = rows 0..7 (lanes 0-15) or
// rows 8..15 (lanes 16-31); one xor-16 shuffle completes the total.
__device__ __forceinline__ float wave_sum_wmma(float v) {
  v2f a; a.x = v; a.y = 0.0f;
  v2f ones; ones.x = 1.0f; ones.y = 1.0f;
  v8f c = {};
  c = __builtin_amdgcn_wmma_f32_16x16x4_f32(false, a, false, ones, (short)0, c,
                                            false, false);
  float s = c[0] + c[1] + c[2] + c[3] + c[4] + c[5] + c[6] + c[7];
  s += __shfl_xor(s, 16, 32);
  return s;  // full-wave total, replicated in every lane
}

__global__ void zero_out(float* out) {
  if (threadIdx.x < 8) out[threadIdx.x] = 0.0f;
}

// One wave32 per image: collect first <=32 GT boxes (flat-index order) with
// obj mask > 0. Matches jax.lax.top_k(mask, 32) tie-break semantics under the
// final max(iou * val).
__global__ void gather_gt(const float* __restrict__ yt, int g,
                          float* __restrict__ gt, int* __restrict__ cnt) {
  const int b = blockIdx.x;
  const int lane = threadIdx.x;  // 32 threads
  const int n = g * g * 3;
  const float* base = yt + (size_t)b * (size_t)n * 85u;
  float* out = gt + (size_t)b * 32u * 8u;
  int count = 0;
  for (int s = 0; s < n && count < 32; s += 32) {
    const int i = s + lane;
    const bool valid = i < n;
    float m = 0.0f;
    if (valid) m = base[(size_t)i * 85u + 4u];
    const bool p = valid && (m > 0.0f);
    const unsigned int bal = __builtin_amdgcn_ballot_w32(p);
    const int pre = __popc(bal & ((1u << lane) - 1u));
    if (p) {
      const int slot = count + pre;
      if (slot < 32) {
        const float* bx = base + (size_t)i * 85u;
        float* o = out + slot * 8;
        o[0] = bx[0]; o[1] = bx[1]; o[2] = bx[2]; o[3] = bx[3]; o[4] = m;
      }
    }
    count += __popc(bal);
  }
  if (lane == 0) cnt[b] = (count < 32) ? count : 32;
}

__global__ __launch_bounds__(256)
void yolo_layer_loss(const float* __restrict__ feats, const float* __restrict__ yt,
                     const float* __restrict__ gt, const int* __restrict__ cnt,
                     int g, float aw0, float ah0, float aw1, float ah1,
                     float aw2, float ah2, float* __restrict__ out) {
  const int gg = g * g;
  const int n = 16 * gg * 3;
  const int idx = blockIdx.x * 256 + threadIdx.x;
  float fxy = 0.f, fwh = 0.f, fconf = 0.f, fcls = 0.f;
  float cR = 0.f, cR75 = 0.f, cP = 0.f;
  if (idx < n) {
    int t = idx;
    const int a  = t % 3; t /= 3;
    const int gx = t % g; t /= g;
    const int gy = t % g;
    const int b  = t / g;
    const float aw = (a == 0) ? aw0 : (a == 1) ? aw1 : aw2;
    const float ah = (a == 0) ? ah0 : (a == 1) ? ah1 : ah2;
    // feats[b][a*85+c][gy][gx]
    const size_t f = ((size_t)b * 255u + (size_t)a * 85u) * (size_t)gg
                   + (size_t)gy * (size_t)g + (size_t)gx;
    const float r0 = feats[f];
    const float r1 = feats[f + (size_t)gg];
    const float r2 = feats[f + 2u * (size_t)gg];
    const float r3 = feats[f + 3u * (size_t)gg];
    const float r4 = feats[f + 4u * (size_t)gg];
    const float s0 = sigm(r0), s1 = sigm(r1);
    const float invg = 1.0f / (float)g;
    const float px = (s0 + (float)gx) * invg;
    const float py = (s1 + (float)gy) * invg;
    const float inv_in = 1.0f / 416.0f;  // input_shape = 13*32 (both dims)
    const float pw = __expf(r2) * aw * inv_in;
    const float ph = __expf(r3) * ah * inv_in;

    // best IoU vs gathered GT boxes of this image
    float best = 0.0f;
    const int c = cnt[b];
    const float* gtb = gt + (size_t)b * 32u * 8u;
    const float parea = pw * ph;
    const float px0 = px - 0.5f * pw, px1 = px + 0.5f * pw;
    const float py0 = py - 0.5f * ph, py1 = py + 0.5f * ph;
    for (int k = 0; k < c; ++k) {
      const float bx = gtb[k * 8 + 0], by = gtb[k * 8 + 1];
      const float bw = gtb[k * 8 + 2], bh = gtb[k * 8 + 3];
      const float val = gtb[k * 8 + 4];
      const float iw = fminf(px1, bx + 0.5f * bw) - fmaxf(px0, bx - 0.5f * bw);
      const float ih = fminf(py1, by + 0.5f * bh) - fmaxf(py0, by - 0.5f * bh);
      const float inter = fmaxf(iw, 0.0f) * fmaxf(ih, 0.0f);
      const float iou = inter / (parea + bw * bh - inter);
      best = fmaxf(best, iou * val);
    }
    const float ignore = (best < 0.5f) ? 1.0f : 0.0f;

    const float* y = yt + (size_t)idx * 85u;
    const float obj = y[4];
    const float tx = y[0] * (float)g - (float)gx;
    const float ty = y[1] * (float)g - (float)gy;
    const float bscale = 2.0f - y[2] * y[3];
    const float ex = obj * (bscale * s0) - obj * tx;
    const float ey = obj * (bscale * s1) - obj * ty;
    fxy = ex * ex + ey * ey;
    const float tw = (obj > 0.0f) ? __logf(y[2] / aw * 416.0f) : 0.0f;
    const float th = (obj > 0.0f) ? __logf(y[3] / ah * 416.0f) : 0.0f;
    fwh = obj * bscale * 0.5f * ((tw - r2) * (tw - r2) + (th - r3) * (th - r3));
    const float sc = sigm(r4);
    fconf = obj * (sc - 1.0f) * (sc - 1.0f) + (1.0f - obj) * ignore * sc * sc;
    float cl = 0.0f;
#pragma unroll 8
    for (int k = 0; k < 80; ++k) {
      const float ss = sigm(feats[f + (size_t)(5 + k) * (size_t)gg]);
      const float d = (ss - y[5 + k]) * obj;
      cl += d * d;
    }
    fcls = cl;
    cR   = (best > 0.5f)  ? 1.0f : 0.0f;
    cR75 = (best > 0.75f) ? 1.0f : 0.0f;
    cP   = (sc > 0.25f)   ? 1.0f : 0.0f;
    const float invm = 1.0f / 16.0f;
    fxy *= invm; fwh *= invm; fconf *= invm; fcls *= invm;
  }
  // Wave32 WMMA reductions. EXEC is all-ones here: no divergent exits above,
  // blockDim is a multiple of 32, tail lanes contribute zeros.
  const float sxy  = wave_sum_wmma(fxy);
  const float swh  = wave_sum_wmma(fwh);
  const float scf  = wave_sum_wmma(fconf);
  const float scl  = wave_sum_wmma(fcls);
  const float sR   = wave_sum_wmma(cR);
  const float sR75 = wave_sum_wmma(cR75);
  const float sP   = wave_sum_wmma(cP);
  if ((threadIdx.x & 31) == 0) {
    atomicAdd(out + 0, sxy + swh + scf + scl);
    atomicAdd(out + 1, sxy);
    atomicAdd(out + 2, swh);
    atomicAdd(out + 3, scf);
    atomicAdd(out + 4, scl);
    atomicAdd(out + 5, sR);
    atomicAdd(out + 6, sR75);
    atomicAdd(out + 7, sP);
  }
}

extern "C" void kernel_launch(void* const* d_in, const int* in_sizes, int n_in,
                              void* d_out, int out_size, void* d_ws, size_t ws_size,
                              hipStream_t stream) {
  (void)in_sizes; (void)n_in; (void)out_size; (void)ws_size;
  // setup_inputs dict order: yolo_out0, y_true0, yolo_out1, y_true1, yolo_out2, y_true2
  const float* feats[3] = {(const float*)d_in[0], (const float*)d_in[2],
                           (const float*)d_in[4]};
  const float* yts[3]   = {(const float*)d_in[1], (const float*)d_in[3],
                           (const float*)d_in[5]};
  float* out = (float*)d_out;
  float* gt  = (float*)d_ws;                                   // 3*16*32*8 f32
  int*   cnt = (int*)((char*)d_ws + 3 * 16 * 32 * 8 * sizeof(float));  // 3*16 i32

  zero_out<<<1, 32, 0, stream>>>(out);

  static const int G[3] = {13, 26, 52};
  // ANCHOR_MASK: layer0 -> anchors {6,7,8}; layer1 -> {3,4,5}; layer2 -> {0,1,2}
  static const float A[3][6] = {
      {116.f, 90.f, 156.f, 198.f, 373.f, 326.f},
      {30.f, 61.f, 62.f, 45.f, 59.f, 119.f},
      {10.f, 13.f, 16.f, 30.f, 33.f, 23.f}};

  for (int l = 0; l < 3; ++l) {
    gather_gt<<<16, 32, 0, stream>>>(yts[l], G[l],
                                     gt + (size_t)l * 16 * 32 * 8, cnt + l * 16);
  }
  for (int l = 0; l < 3; ++l) {
    const int n = 16 * G[l] * G[l] * 3;
    const int blocks = (n + 255) / 256;
    yolo_layer_loss<<<blocks, 256, 0, stream>>>(
        feats[l], yts[l], gt + (size_t)l * 16 * 32 * 8, cnt + l * 16, G[l],
        A[l][0], A[l][1], A[l][2], A[l][3], A[l][4], A[l][5], out);
  }
}